// GCN_encoder_scatter_78520592105494
// MI455X (gfx1250) — compile-verified
//
#include <hip/hip_runtime.h>
#include <hip/hip_bf16.h>

typedef __attribute__((ext_vector_type(2))) float v2f;
typedef __attribute__((ext_vector_type(8))) float v8f;

#define FDIM 128   // F == H == 128

// ---------------------------------------------------------------------------
// deg[i] = 1.0 (self loop)
__global__ void k_deg_init(float* __restrict__ deg, int n) {
    int i = blockIdx.x * blockDim.x + threadIdx.x;
    if (i < n) deg[i] = 1.0f;
}

// deg[col[e]] += 1
__global__ void k_deg_count(const int* __restrict__ col, float* __restrict__ deg, int e) {
    int i = blockIdx.x * blockDim.x + threadIdx.x;
    if (i < e) atomicAdd(&deg[col[i]], 1.0f);
}

// deg -> deg^{-1/2} in place
__global__ void k_deg_rsqrt(float* __restrict__ deg, int n) {
    int i = blockIdx.x * blockDim.x + threadIdx.x;
    if (i < n) deg[i] = rsqrtf(deg[i]);
}

// ---------------------------------------------------------------------------
// h = x @ W^T via V_WMMA_F32_16X16X4_F32.
// One block per 16-node tile. A-tile (16x128 f32, 8KB) staged in LDS, shared
// by 8 waves; wave w computes the 16x16 output tile for H-cols [16w, 16w+16).
// Fragment layouts per CDNA5 ISA 7.12.2:
//   A 16x4 f32 : lane l -> m = l%16, VGPR0 = A[m][k+2*(l/16)], VGPR1 = next k
//   B 4x16 f32 : lane l -> n = l%16, VGPR i = B[i+2*(l/16)][n]
//   C/D 16x16  : VGPR i, lane l -> row (i + 8*(l/16)), col l%16
__global__ __launch_bounds__(256) void k_gemm_wmma(const float* __restrict__ x,
                                                   const float* __restrict__ W,
                                                   float* __restrict__ h, int n) {
    __shared__ float As[16 * FDIM];

    const long long nbase = (long long)blockIdx.x * 16;
    const long long xcap  = (long long)n * FDIM;

    // cooperative A-tile load (2048 floats, 8 per thread, coalesced)
    const float* xsrc = x + nbase * FDIM;
    for (int i = threadIdx.x; i < 16 * FDIM; i += 256)
        As[i] = (nbase * FDIM + i < xcap) ? xsrc[i] : 0.0f;
    __syncthreads();

    const int wid  = threadIdx.x >> 5;   // 0..7 -> H tile
    const int lane = threadIdx.x & 31;
    const int grp  = lane >> 4;          // 0/1: K-pair / row-half selector
    const int m    = lane & 15;          // row within tile (also B's n)
    const int hbase = wid * 16;

    const float* Wrow = W + (long long)(hbase + m) * FDIM;  // B[k][n] = W[hbase+n][k]
    const float* Arow = &As[m * FDIM];

    v8f c = {};
    for (int k = 0; k < FDIM; k += 4) {
        v2f a, b;
        a.x = Arow[k + 2 * grp];
        a.y = Arow[k + 2 * grp + 1];
        b.x = Wrow[k + 2 * grp];
        b.y = Wrow[k + 2 * grp + 1];
        // (neg_a, A, neg_b, B, c_mod, C, reuse_a, reuse_b)
        c = __builtin_amdgcn_wmma_f32_16x16x4_f32(false, a, false, b,
                                                  (short)0, c, false, false);
    }

    for (int i = 0; i < 8; ++i) {
        long long row = nbase + i + 8 * grp;
        if (row < n) h[row * FDIM + hbase + m] = c[i];
    }
}

// ---------------------------------------------------------------------------
// out[i][:] = bias[:] + dinv[i]^2 * h[i][:]   (self-loop term, atomic-free)
// one wave per node, float4 per lane
__global__ __launch_bounds__(256) void k_out_init(const float* __restrict__ h,
                                                  const float* __restrict__ dinv,
                                                  const float* __restrict__ bias,
                                                  float* __restrict__ out, int n) {
    int node = blockIdx.x * (blockDim.x >> 5) + (threadIdx.x >> 5);
    if (node >= n) return;
    int lane = threadIdx.x & 31;
    float w = dinv[node];
    w = w * w;
    const float4 hv = ((const float4*)(h + (long long)node * FDIM))[lane];
    const float4 bv = ((const float4*)bias)[lane];
    float4 o;
    o.x = bv.x + w * hv.x;
    o.y = bv.y + w * hv.y;
    o.z = bv.z + w * hv.z;
    o.w = bv.w + w * hv.w;
    ((float4*)(out + (long long)node * FDIM))[lane] = o;
}

// ---------------------------------------------------------------------------
// out[col[e]][:] += dinv[row]*dinv[col] * h[row[e]][:]
// one wave per edge: coalesced float4 gather + 4 f32 atomics per lane (RMW in L2)
__global__ __launch_bounds__(256) void k_scatter(const float* __restrict__ h,
                                                 const int* __restrict__ row,
                                                 const int* __restrict__ col,
                                                 const float* __restrict__ dinv,
                                                 float* __restrict__ out, int e) {
    int edge = blockIdx.x * (blockDim.x >> 5) + (threadIdx.x >> 5);
    if (edge >= e) return;
    int lane = threadIdx.x & 31;
    int r = row[edge];
    int c = col[edge];
    float w = dinv[r] * dinv[c];
    const float4 v = ((const float4*)(h + (long long)r * FDIM))[lane];
    float* op = out + (long long)c * FDIM + lane * 4;
    atomicAdd(op + 0, w * v.x);
    atomicAdd(op + 1, w * v.y);
    atomicAdd(op + 2, w * v.z);
    atomicAdd(op + 3, w * v.w);
}

// ---------------------------------------------------------------------------
extern "C" void kernel_launch(void* const* d_in, const int* in_sizes, int n_in,
                              void* d_out, int out_size, void* d_ws, size_t ws_size,
                              hipStream_t stream) {
    const float* x    = (const float*)d_in[0];   // [N, F]
    const int*   ei   = (const int*)d_in[1];     // [2, E]
    const float* W    = (const float*)d_in[3];   // [H, F]
    const float* bias = (const float*)d_in[4];   // [H]
    float* out        = (float*)d_out;           // [N, H]

    const int N = in_sizes[0] / FDIM;
    const int E = in_sizes[1] / 2;
    const int* row = ei;
    const int* col = ei + E;

    // workspace: h [N*H] floats, then deg/dinv [N] floats
    float* h   = (float*)d_ws;
    float* deg = h + (size_t)N * FDIM;

    // 1) degree with self-loops, then deg^{-1/2} in place
    k_deg_init <<<(N + 255) / 256, 256, 0, stream>>>(deg, N);
    k_deg_count<<<(E + 255) / 256, 256, 0, stream>>>(col, deg, E);
    k_deg_rsqrt<<<(N + 255) / 256, 256, 0, stream>>>(deg, N);

    // 2) h = x @ W^T  (fp32 WMMA)
    k_gemm_wmma<<<(N + 15) / 16, 256, 0, stream>>>(x, W, h, N);

    // 3) out = bias + dinv^2 * h      (self-loop contribution)
    k_out_init<<<(N + 7) / 8, 256, 0, stream>>>(h, deg, bias, out, N);

    // 4) edge scatter with f32 atomics
    k_scatter<<<(E + 7) / 8, 256, 0, stream>>>(h, row, col, deg, out, E);
}